// NodeEdgeLayer_50869592655492
// MI455X (gfx1250) — compile-verified
//
#include <hip/hip_runtime.h>

typedef __attribute__((ext_vector_type(2))) float v2f;
typedef __attribute__((ext_vector_type(8))) float v8f;

static constexpr int HD = 128;   // hidden dim

// ---------------------------------------------------------------------------
// Stage a 128x128 block of a row-major weight matrix (row stride = 128,
// starting at row r0) into LDS, pre-swizzled into the B-fragment layout of
// V_WMMA_F32_16X16X4_F32:
//   fragment (chunk, ntile) holds B[k0+kin][n0+lo] where k0=4*chunk, n0=16*nt,
//   lane = hh*16+lo holds VGPR j = B[k0 + 2*hh + j][n0 + lo]  (j in {0,1})
// stored so each lane reads a contiguous v2f -> conflict-free ds_load_b64.
// ---------------------------------------------------------------------------
__device__ __forceinline__ void stage_w_swizzled(const float* __restrict__ W,
                                                 int r0, float* sW) {
  for (int idx = threadIdx.x; idx < 128 * 128; idx += blockDim.x) {
    int klocal = idx >> 7;       // 0..127
    int n      = idx & 127;      // 0..127
    int chunk  = klocal >> 2;    // 0..31
    int kin    = klocal & 3;     // 0..3
    int j      = kin & 1;
    int hh     = kin >> 1;
    int lo     = n & 15;
    int nt     = n >> 4;
    int lane   = hh * 16 + lo;
    sW[((chunk << 3) + nt) * 64 + lane * 2 + j] =
        W[(size_t)(r0 + klocal) * HD + n];
  }
}

__device__ __forceinline__ v2f ldv2(const float* p) { return *(const v2f*)p; }

// ---------------------------------------------------------------------------
// Kernel 1: per 16-edge tile:
//   A1 = [node_rep[src]+node_rep[dst] | edge_rep]  (16 x 256)
//   edge_h = relu(A1 @ W1)            -> edge region of d_out (scratch reuse)
//   lvl_aggr[src] += edge_h ; lvl_aggr[dst] += edge_h  (L2-resident atomics)
// ---------------------------------------------------------------------------
__global__ __launch_bounds__(256) void edge_mlp1_kernel(
    const float* __restrict__ node_rep, const float* __restrict__ edge_rep,
    const int* __restrict__ src, const int* __restrict__ dst,
    const float* __restrict__ W1, float* __restrict__ edge_h,
    float* __restrict__ lvl, int E) {
  __shared__ float sW[32 * 8 * 64];  // 64 KB swizzled B fragments
  const int lane = threadIdx.x & 31;
  const int wave = threadIdx.x >> 5;
  const int tile = blockIdx.x * 8 + wave;
  const int e0 = tile * 16;
  const int hh = lane >> 4;
  const int lo = lane & 15;

  const int em = min(e0 + lo, E - 1);   // this lane's A-row edge
  const int srow = src[em];
  const int drow = dst[em];

  v8f acc[8] = {};
  for (int khalf = 0; khalf < 2; ++khalf) {
    __syncthreads();
    stage_w_swizzled(W1, khalf * 128, sW);
    __syncthreads();
    if (e0 < E) {  // wave-uniform: EXEC all-ones inside (WMMA requirement)
#pragma unroll 4
      for (int chunk = 0; chunk < 32; ++chunk) {
        const int kg = chunk * 4 + hh * 2;  // K offset within this half
        v2f a;
        if (khalf == 0) {
          a = ldv2(node_rep + (size_t)srow * HD + kg) +
              ldv2(node_rep + (size_t)drow * HD + kg);
        } else {
          a = ldv2(edge_rep + (size_t)em * HD + kg);
        }
#pragma unroll
        for (int nt = 0; nt < 8; ++nt) {
          v2f b = *(const v2f*)&sW[((chunk << 3) + nt) * 64 + lane * 2];
          acc[nt] = __builtin_amdgcn_wmma_f32_16x16x4_f32(
              false, a, false, b, (short)0, acc[nt], false, false);
        }
      }
    }
  }
  if (e0 >= E) return;

#pragma unroll
  for (int nt = 0; nt < 8; ++nt) {
#pragma unroll
    for (int r = 0; r < 8; ++r) {
      const int M = r + 8 * hh;      // D layout: M = vgpr + 8*(lane/16)
      const int e = e0 + M;
      if (e < E) {
        const float v = fmaxf(acc[nt][r], 0.0f);
        const int n = nt * 16 + lo;  // D layout: N = lane%16 within ntile
        edge_h[(size_t)e * HD + n] = v;
        atomicAdd(&lvl[(size_t)src[e] * HD + n], v);
        atomicAdd(&lvl[(size_t)dst[e] * HD + n], v);
      }
    }
  }
}

// ---------------------------------------------------------------------------
// Kernel 2: node_out = relu(((1+eps1)*node_rep + lvl_aggr) @ W2)
// ---------------------------------------------------------------------------
__global__ __launch_bounds__(256) void node_mlp_kernel(
    const float* __restrict__ node_rep, const float* __restrict__ lvl,
    const float* __restrict__ W2, const float* __restrict__ eps1,
    float* __restrict__ node_out, int N) {
  __shared__ float sW[32 * 8 * 64];
  __syncthreads();
  stage_w_swizzled(W2, 0, sW);
  __syncthreads();

  const int lane = threadIdx.x & 31;
  const int wave = threadIdx.x >> 5;
  const int tile = blockIdx.x * 8 + wave;
  const int r0 = tile * 16;
  const int hh = lane >> 4;
  const int lo = lane & 15;
  const float s1 = 1.0f + eps1[0];

  const int rm = min(r0 + lo, N - 1);
  v8f acc[8] = {};
  if (r0 < N) {  // wave-uniform
#pragma unroll 4
    for (int chunk = 0; chunk < 32; ++chunk) {
      const int kg = chunk * 4 + hh * 2;
      v2f a = ldv2(node_rep + (size_t)rm * HD + kg) * s1 +
              ldv2(lvl + (size_t)rm * HD + kg);
#pragma unroll
      for (int nt = 0; nt < 8; ++nt) {
        v2f b = *(const v2f*)&sW[((chunk << 3) + nt) * 64 + lane * 2];
        acc[nt] = __builtin_amdgcn_wmma_f32_16x16x4_f32(
            false, a, false, b, (short)0, acc[nt], false, false);
      }
    }
  }
  if (r0 >= N) return;
#pragma unroll
  for (int nt = 0; nt < 8; ++nt) {
#pragma unroll
    for (int r = 0; r < 8; ++r) {
      const int row = r0 + r + 8 * hh;
      if (row < N) {
        node_out[(size_t)row * HD + nt * 16 + lo] = fmaxf(acc[nt][r], 0.0f);
      }
    }
  }
}

// ---------------------------------------------------------------------------
// Kernel 3: edge_out = relu(((1+eps2)*edge_h + lift_aggr) @ W_lift)
// edge_h is read from (and edge_out written back to) the edge region of
// d_out, in place; lift_aggr is re-gathered from L2-resident node_rep.
// ---------------------------------------------------------------------------
__global__ __launch_bounds__(256) void edge_mlp2_kernel(
    const float* __restrict__ node_rep, const int* __restrict__ src,
    const int* __restrict__ dst, const float* __restrict__ W_lift,
    const float* __restrict__ eps2, float* __restrict__ edge_io, int E) {
  __shared__ float sW[32 * 8 * 64];
  __syncthreads();
  stage_w_swizzled(W_lift, 0, sW);
  __syncthreads();

  const int lane = threadIdx.x & 31;
  const int wave = threadIdx.x >> 5;
  const int tile = blockIdx.x * 8 + wave;
  const int e0 = tile * 16;
  const int hh = lane >> 4;
  const int lo = lane & 15;
  const float s2 = 1.0f + eps2[0];

  const int em = min(e0 + lo, E - 1);
  const int srow = src[em];
  const int drow = dst[em];

  v8f acc[8] = {};
  if (e0 < E) {  // wave-uniform
#pragma unroll 4
    for (int chunk = 0; chunk < 32; ++chunk) {
      const int kg = chunk * 4 + hh * 2;
      v2f lift = ldv2(node_rep + (size_t)srow * HD + kg) +
                 ldv2(node_rep + (size_t)drow * HD + kg);
      v2f a = ldv2(edge_io + (size_t)em * HD + kg) * s2 + lift;
#pragma unroll
      for (int nt = 0; nt < 8; ++nt) {
        v2f b = *(const v2f*)&sW[((chunk << 3) + nt) * 64 + lane * 2];
        acc[nt] = __builtin_amdgcn_wmma_f32_16x16x4_f32(
            false, a, false, b, (short)0, acc[nt], false, false);
      }
    }
  }
  if (e0 >= E) return;
#pragma unroll
  for (int nt = 0; nt < 8; ++nt) {
#pragma unroll
    for (int r = 0; r < 8; ++r) {
      const int e = e0 + r + 8 * hh;
      if (e < E) {
        edge_io[(size_t)e * HD + nt * 16 + lo] = fmaxf(acc[nt][r], 0.0f);
      }
    }
  }
}

// ---------------------------------------------------------------------------
extern "C" void kernel_launch(void* const* d_in, const int* in_sizes, int n_in,
                              void* d_out, int out_size, void* d_ws,
                              size_t ws_size, hipStream_t stream) {
  (void)n_in; (void)out_size; (void)ws_size;
  const float* node_rep = (const float*)d_in[0];
  const float* edge_rep = (const float*)d_in[1];
  const int*   eidx     = (const int*)d_in[2];
  const float* W1       = (const float*)d_in[3];
  const float* W2       = (const float*)d_in[4];
  const float* W_lift   = (const float*)d_in[5];
  const float* eps1     = (const float*)d_in[6];
  const float* eps2     = (const float*)d_in[7];

  const int N = in_sizes[0] / HD;
  const int E = in_sizes[1] / HD;
  const int* src = eidx;
  const int* dst = eidx + E;

  float* node_out = (float*)d_out;                     // [N,128]
  float* edge_io  = (float*)d_out + (size_t)N * HD;    // [E,128] scratch->out
  float* lvl      = (float*)d_ws;                      // [N,128] scatter acc

  hipMemsetAsync(lvl, 0, (size_t)N * HD * sizeof(float), stream);

  const int etiles  = (E + 15) / 16;
  const int eblocks = (etiles + 7) / 8;
  const int ntiles  = (N + 15) / 16;
  const int nblocks = (ntiles + 7) / 8;

  edge_mlp1_kernel<<<eblocks, 256, 0, stream>>>(node_rep, edge_rep, src, dst,
                                                W1, edge_io, lvl, E);
  node_mlp_kernel<<<nblocks, 256, 0, stream>>>(node_rep, lvl, W2, eps1,
                                               node_out, N);
  edge_mlp2_kernel<<<eblocks, 256, 0, stream>>>(node_rep, src, dst, W_lift,
                                                eps2, edge_io, E);
}